// KVKWCache_35021163331636
// MI455X (gfx1250) — compile-verified
//
#include <hip/hip_runtime.h>
#include <cstdint>
#include <cstddef>

// Problem constants (from reference): B=8, H=16, S=4096, D=128, s=4
#define BQ 8
#define HQ 16
#define SQ 4096
#define DQ 128
#define NS 4
#define KCHUNK 4   // async b128 transfers per lane per pipeline phase

// ---------------------------------------------------------------------------
// Streaming per-batch gather copy:  dst[b'] = src[batch_indexes[b']]
// CDNA5 async global<->LDS 128-bit transfers (ASYNCcnt-tracked), software
// pipelined: two sets of KCHUNK LDS slots; each phase overlaps the stores of
// set A with the loads of set B, keeping up to 2*KCHUNK async ops (~4 KB/wave)
// in flight between s_wait_asynccnt barriers.
// Dynamic LDS (no static LDS) => our allocation starts at LDS byte offset 0.
// ---------------------------------------------------------------------------
__global__ void gather_copy_async(const float* __restrict__ src,
                                  float* __restrict__ dst,
                                  const int* __restrict__ batch_indexes,
                                  size_t batch_elems /* floats per batch */) {
  extern __shared__ char dynlds[];  // 2 * KCHUNK * blockDim.x * 16 bytes
  (void)dynlds;

  const int b = blockIdx.y;
  const float* sbase = src + (size_t)batch_indexes[b] * batch_elems;
  float* dbase = dst + (size_t)b * batch_elems;

  const size_t nvec   = batch_elems >> 2;                 // float4 count
  const size_t stride = (size_t)gridDim.x * blockDim.x;   // float4 step
  size_t i = (size_t)blockIdx.x * blockDim.x + threadIdx.x;

  const unsigned laneSlot  = threadIdx.x * 16u;
  const unsigned slotPitch = (unsigned)blockDim.x * 16u;  // bytes per slot row
  unsigned baseA = 0u;
  unsigned baseB = (unsigned)KCHUNK * slotPitch;

  // ---- preload set A ----
  int cntA = 0;
  size_t iA = i;
#pragma unroll
  for (int k = 0; k < KCHUNK; ++k) {
    if (i < nvec) {
      const unsigned long long g = (unsigned long long)(sbase + (i << 2));
      const unsigned lds = baseA + (unsigned)k * slotPitch + laneSlot;
      asm volatile("global_load_async_to_lds_b128 %0, %1, off th:TH_LOAD_NT"
                   :: "v"(lds), "v"(g) : "memory");
      ++cntA;
      i += stride;
    }
  }

  while (cntA > 0) {
    // Everything outstanding (loads of current set, stores of the other set)
    // is complete after this: current set's data is valid, other set is free.
    asm volatile("s_wait_asynccnt 0x0" ::: "memory");

    // ---- issue loads for next set (overlaps with the stores below) ----
    int cntB = 0;
    size_t iB = i;
#pragma unroll
    for (int k = 0; k < KCHUNK; ++k) {
      if (i < nvec) {
        const unsigned long long g = (unsigned long long)(sbase + (i << 2));
        const unsigned lds = baseB + (unsigned)k * slotPitch + laneSlot;
        asm volatile("global_load_async_to_lds_b128 %0, %1, off th:TH_LOAD_NT"
                     :: "v"(lds), "v"(g) : "memory");
        ++cntB;
        i += stride;
      }
    }

    // ---- store current set ----
    size_t j = iA;
#pragma unroll
    for (int k = 0; k < KCHUNK; ++k) {
      if (k < cntA) {
        const unsigned long long d = (unsigned long long)(dbase + (j << 2));
        const unsigned lds = baseA + (unsigned)k * slotPitch + laneSlot;
        asm volatile("global_store_async_from_lds_b128 %0, %1, off th:TH_STORE_NT"
                     :: "v"(d), "v"(lds) : "memory");
        j += stride;
      }
    }

    const unsigned t = baseA; baseA = baseB; baseB = t;
    cntA = cntB;
    iA = iB;
  }
  asm volatile("s_wait_asynccnt 0x0" ::: "memory");
}

// ---------------------------------------------------------------------------
// k/v scatter: out[p, h, pos(b,j), :] = val[b, h, j, :]
// for every output batch p with batch_indexes[p] == batch_indexes[b].
// ---------------------------------------------------------------------------
__global__ void kv_scatter(const float* __restrict__ k_val,
                           const float* __restrict__ v_val,
                           const int* __restrict__ input_pos,
                           const int* __restrict__ batch_indexes,
                           float* __restrict__ out_k,
                           float* __restrict__ out_v) {
  const unsigned idx = blockIdx.x * blockDim.x + threadIdx.x;  // < B*H*s*D
  const unsigned d = idx & (DQ - 1);
  const unsigned j = (idx >> 7) & (NS - 1);
  const unsigned h = (idx >> 9) & (HQ - 1);
  const unsigned b = idx >> 13;
  if (b >= BQ) return;

  int pos = input_pos[b * NS + j] % SQ;
  if (pos < 0) pos += SQ;

  const size_t vidx = (((size_t)b * HQ + h) * NS + j) * DQ + d;
  const float kv = k_val[vidx];
  const float vv = v_val[vidx];
  const int target = batch_indexes[b];

  for (int p = 0; p < BQ; ++p) {
    if (batch_indexes[p] == target) {
      const size_t o = (((size_t)p * HQ + h) * SQ + (unsigned)pos) * DQ + d;
      out_k[o] = kv;
      out_v[o] = vv;
    }
  }
}

// ---------------------------------------------------------------------------
// Generic row scatter for kw (R = 2*H*H = 512) and kw_sub (R = 5*2*H = 160):
// out[p, pos(b,j), <R contiguous floats>] = val[b, j, <R floats>]
// ---------------------------------------------------------------------------
__global__ void row_scatter(const float* __restrict__ val,
                            float* __restrict__ out,
                            const int* __restrict__ input_pos,
                            const int* __restrict__ batch_indexes,
                            int R, int total) {
  const int idx = blockIdx.x * blockDim.x + threadIdx.x;
  if (idx >= total) return;
  const int r  = idx % R;
  const int bj = idx / R;
  const int j  = bj % NS;
  const int b  = bj / NS;

  int pos = input_pos[b * NS + j] % SQ;
  if (pos < 0) pos += SQ;

  const float v = val[idx];  // val is (B, s, R) contiguous -> flat == idx
  const int target = batch_indexes[b];
  for (int p = 0; p < BQ; ++p) {
    if (batch_indexes[p] == target) {
      out[((size_t)p * SQ + (unsigned)pos) * (size_t)R + r] = v;
    }
  }
}

// ---------------------------------------------------------------------------
extern "C" void kernel_launch(void* const* d_in, const int* in_sizes, int n_in,
                              void* d_out, int out_size, void* d_ws, size_t ws_size,
                              hipStream_t stream) {
  (void)in_sizes; (void)n_in; (void)out_size; (void)d_ws; (void)ws_size;

  const float* k_cache      = (const float*)d_in[0];
  const float* v_cache      = (const float*)d_in[1];
  const float* kw_cache     = (const float*)d_in[2];
  const float* kw_sub_cache = (const float*)d_in[3];
  const int*   input_pos    = (const int*)d_in[4];
  const int*   batch_idx    = (const int*)d_in[5];
  const float* k_val        = (const float*)d_in[6];
  const float* v_val        = (const float*)d_in[7];
  const float* kw_val       = (const float*)d_in[8];
  const float* kw_sub       = (const float*)d_in[9];

  const size_t kvBatch = (size_t)HQ * SQ * DQ;        // 8,388,608 floats
  const size_t kwBatch = (size_t)SQ * 2 * HQ * HQ;    // 2,097,152 floats
  const size_t ksBatch = (size_t)SQ * 5 * 2 * HQ;     //   655,360 floats

  float* out    = (float*)d_out;
  float* out_k  = out;
  float* out_v  = out_k  + (size_t)BQ * kvBatch;
  float* out_kw = out_v  + (size_t)BQ * kvBatch;
  float* out_ks = out_kw + (size_t)BQ * kwBatch;

  const int THREADS = 256;
  const size_t LDS_BYTES = 2u * KCHUNK * THREADS * 16u;  // 32 KB: 2 slot sets

  auto launch_copy = [&](const float* s, float* d, size_t batch_elems) {
    const size_t nvec = batch_elems >> 2;
    size_t bx = (nvec + THREADS - 1) / THREADS;
    if (bx > 512) bx = 512;
    dim3 grid((unsigned)bx, BQ);
    gather_copy_async<<<grid, THREADS, LDS_BYTES, stream>>>(s, d, batch_idx,
                                                            batch_elems);
  };

  // Phase 1: bandwidth-bound gathered full-cache copies (~1.17 GB traffic).
  launch_copy(k_cache,      out_k,  kvBatch);
  launch_copy(v_cache,      out_v,  kvBatch);
  launch_copy(kw_cache,     out_kw, kwBatch);
  launch_copy(kw_sub_cache, out_ks, ksBatch);

  // Phase 2: tiny scatter overwrites (stream-ordered after the copies).
  {
    const int n = BQ * HQ * NS * DQ;  // 65536
    kv_scatter<<<(n + THREADS - 1) / THREADS, THREADS, 0, stream>>>(
        k_val, v_val, input_pos, batch_idx, out_k, out_v);
  }
  {
    const int R = 2 * HQ * HQ;        // 512
    const int n = BQ * NS * R;        // 16384
    row_scatter<<<(n + THREADS - 1) / THREADS, THREADS, 0, stream>>>(
        kw_val, out_kw, input_pos, batch_idx, R, n);
  }
  {
    const int R = 5 * 2 * HQ;         // 160
    const int n = BQ * NS * R;        // 5120
    row_scatter<<<(n + THREADS - 1) / THREADS, THREADS, 0, stream>>>(
        kw_sub, out_ks, input_pos, batch_idx, R, n);
  }
}